// GlobalAttn_38654705664064
// MI455X (gfx1250) — compile-verified
//
#include <hip/hip_runtime.h>
#include <math.h>

typedef __attribute__((ext_vector_type(2))) float v2f;
typedef __attribute__((ext_vector_type(8))) float v8f;

// ---------------------------------------------------------------------------
// Small-M f32 WMMA GEMM:  C[M,N] = A[M,K] * B[K,N]   (all row-major, no bias)
// One wave32 per 16x16 output tile. grid = (N/16, M/16), block = 32.
// Uses V_WMMA_F32_16X16X4_F32 (exact f32 -> matches f32 reference numerics).
// ---------------------------------------------------------------------------
__global__ __launch_bounds__(32) void wmma_gemm_f32(const float* __restrict__ A,
                                                    const float* __restrict__ B,
                                                    float* __restrict__ C,
                                                    int N, int K) {
    const int n0   = blockIdx.x << 4;
    const int m0   = blockIdx.y << 4;
    const int lane = threadIdx.x & 31;
    const int r    = lane & 15;     // row/col within tile
    const int hlf  = lane >> 4;     // which K-pair this lane holds

    v8f acc = {};
    const float* __restrict__ arow = A + (size_t)(m0 + r) * K;
    const float* __restrict__ bcol = B + (n0 + r);

    #pragma unroll 4
    for (int k = 0; k < K; k += 4) {
        v2f a, b;
        // A 16x4 fragment: lanes 0-15 -> K = k+0,k+1 ; lanes 16-31 -> K = k+2,k+3
        a.x = arow[k + 2 * hlf];
        a.y = arow[k + 2 * hlf + 1];
        // B 4x16 fragment (column n0+r), same K split
        b.x = bcol[(size_t)(k + 2 * hlf)     * N];
        b.y = bcol[(size_t)(k + 2 * hlf + 1) * N];
        acc = __builtin_amdgcn_wmma_f32_16x16x4_f32(
                  false, a, false, b, (short)0, acc, false, false);
    }

    // D layout: VGPR i -> row m0+i (lanes 0-15) / m0+i+8 (lanes 16-31), col n0+r
    #pragma unroll
    for (int i = 0; i < 8; ++i)
        C[(size_t)(m0 + i + 8 * hlf) * N + n0 + r] = acc[i];
}

// ---------------------------------------------------------------------------
// scores[b, l] = u[b, :] . enc[l, b, :]   -- the single 512 MiB streaming pass.
// rows are (l*B + b); one wave per row (8 KiB contiguous), float4 loads,
// wave32 butterfly reduction. 8 waves / 256-thread block.
// ---------------------------------------------------------------------------
__global__ __launch_bounds__(256) void score_kernel(const float* __restrict__ enc, // [L*B, K2]
                                                    const float* __restrict__ u,   // [B,  K2]
                                                    float* __restrict__ scores,    // [B,  L]
                                                    int Bsz, int L, int K2) {
    const int wave = threadIdx.x >> 5;
    const int lane = threadIdx.x & 31;
    const long long row = (long long)blockIdx.x * 8 + wave;  // row = l*Bsz + b
    const int b = (int)(row % Bsz);
    const int l = (int)(row / Bsz);

    const float4* __restrict__ e4 = (const float4*)(enc + (size_t)row * K2);
    const float4* __restrict__ u4 = (const float4*)(u + (size_t)b * K2);

    float acc = 0.f;
    const int nv = K2 >> 2;                    // 512 float4's per row
    for (int i = lane; i < nv; i += 32) {
        float4 e = e4[i];
        float4 w = u4[i];
        acc = fmaf(e.x, w.x, acc);
        acc = fmaf(e.y, w.y, acc);
        acc = fmaf(e.z, w.z, acc);
        acc = fmaf(e.w, w.w, acc);
    }
    #pragma unroll
    for (int off = 16; off > 0; off >>= 1)
        acc += __shfl_xor(acc, off, 32);

    if (lane == 0)
        scores[(size_t)b * L + l] = acc;
}

// ---------------------------------------------------------------------------
// Row softmax over L, one 256-thread block (8 wave32) per batch row b.
// ---------------------------------------------------------------------------
__global__ __launch_bounds__(256) void softmax_kernel(const float* __restrict__ scores,
                                                      float* __restrict__ out, int L) {
    __shared__ float red[8];
    const int t = threadIdx.x;
    const float* __restrict__ srow = scores + (size_t)blockIdx.x * L;
    float* __restrict__ orow       = out    + (size_t)blockIdx.x * L;

    float m = -INFINITY;
    for (int i = t; i < L; i += 256) m = fmaxf(m, srow[i]);
    #pragma unroll
    for (int off = 16; off > 0; off >>= 1) m = fmaxf(m, __shfl_xor(m, off, 32));
    if ((t & 31) == 0) red[t >> 5] = m;
    __syncthreads();
    m = red[0];
    #pragma unroll
    for (int w = 1; w < 8; ++w) m = fmaxf(m, red[w]);
    __syncthreads();

    float s = 0.f;
    for (int i = t; i < L; i += 256) {
        float e = __expf(srow[i] - m);
        orow[i] = e;
        s += e;
    }
    #pragma unroll
    for (int off = 16; off > 0; off >>= 1) s += __shfl_xor(s, off, 32);
    if ((t & 31) == 0) red[t >> 5] = s;
    __syncthreads();
    s = 0.f;
    #pragma unroll
    for (int w = 0; w < 8; ++w) s += red[w];
    const float inv = 1.f / s;
    for (int i = t; i < L; i += 256) orow[i] *= inv;
}

// ---------------------------------------------------------------------------
extern "C" void kernel_launch(void* const* d_in, const int* in_sizes, int n_in,
                              void* d_out, int out_size, void* d_ws, size_t ws_size,
                              hipStream_t stream) {
    (void)in_sizes; (void)n_in; (void)out_size; (void)ws_size;

    constexpr int B  = 32;
    constexpr int L  = 2048;
    constexpr int H  = 1024;
    constexpr int K2 = 2 * H;   // 2048

    const float* hidden   = (const float*)d_in[0];  // [B, H]
    const float* enc      = (const float*)d_in[1];  // [L, B, 2H]
    const float* reduce_W = (const float*)d_in[2];  // [H, 2H]
    // d_in[3] reduce_b, d_in[5] attn_b: their score contribution is constant
    // per-b and cancels in the softmax over l -> unused.
    const float* attn_W   = (const float*)d_in[4];  // [H, H]
    float* out            = (float*)d_out;          // [B, L]

    float* v      = (float*)d_ws;                   // [B, H]   128 KiB
    float* u      = v + (size_t)B * H;              // [B, 2H]  256 KiB
    float* scores = u + (size_t)B * K2;             // [B, L]   256 KiB

    // 1) v[b,h] = sum_g hidden[b,g] * attn_W[g,h]   (M=32, N=1024, K=1024)
    wmma_gemm_f32<<<dim3(H / 16, B / 16), 32, 0, stream>>>(hidden, attn_W, v, H, H);

    // 2) u[b,k] = sum_h v[b,h] * reduce_W[h,k]      (M=32, N=2048, K=1024)
    wmma_gemm_f32<<<dim3(K2 / 16, B / 16), 32, 0, stream>>>(v, reduce_W, u, K2, H);

    // 3) scores[b,l] = u[b] . enc[l,b]  -- memory-bound 512 MiB stream
    score_kernel<<<(L * B) / 8, 256, 0, stream>>>(enc, u, scores, B, L, K2);

    // 4) softmax over L per batch row
    softmax_kernel<<<B, 256, 0, stream>>>(scores, out, L);
}